// Wavelet_17643725652465
// MI455X (gfx1250) — compile-verified
//
#include <hip/hip_runtime.h>
#include <cstdint>
#include <cstddef>

// ---------------------------------------------------------------------------
// 5-level Haar wavedec on (64,64,8192) fp32  ==  per-32-block y = W(32x32) x,
// W constant, followed by band-strided scatter. CDNA5 path:
//   - global_load_async_to_lds_b128 staging, double-buffered, pipelined via
//     s_wait_asynccnt 4 / 0 (async loads complete in order)
//   - 32x v_wmma_f32_16x16x4_f32 per wave-task pair (fp32 == reference dtype)
//   - W fragments constexpr-baked into __constant__ memory in exact per-lane
//     fragment order (coalesced b64 loads, no runtime cndmask chains)
//   - scatter closed-formed per half-wave -> b64/b128 stores
// Memory-bound: 256 MiB @ 23.3 TB/s => ~11.5 us floor; WMMA work negligible.
// ---------------------------------------------------------------------------

typedef float v2f __attribute__((ext_vector_type(2)));
typedef float v8f __attribute__((ext_vector_type(8)));

#if defined(__has_builtin)
#  if __has_builtin(__builtin_amdgcn_global_load_async_to_lds_b128) && \
      __has_builtin(__builtin_amdgcn_s_wait_asynccnt)
#    define HAVE_ASYNC_LDS 1
#  endif
#endif
#ifndef HAVE_ASYNC_LDS
#  define HAVE_ASYNC_LDS 0
#endif

#if HAVE_ASYNC_LDS
typedef int b128_t __attribute__((vector_size(16)));
typedef __attribute__((address_space(1))) b128_t* g_b128_ptr;
typedef __attribute__((address_space(3))) b128_t* l_b128_ptr;
#endif

#define LDS_STRIDE 36                 // 32 floats + 4 pad: 16B rows, no conflicts
#define SLICE      (16 * LDS_STRIDE)  // one task's 16 blocks, padded (576 f)

// ---- Compile-time W[r][c] of the 32x32 block-diagonal Haar analysis matrix.
// Row order: [cA5, cD5, cD4(2), cD3(4), cD2(8), cD1(16)].
constexpr float haar_w_c(int r, int c) {
  constexpr float m1 = 0.70710678118654752f;   // 2^-0.5
  constexpr float m2 = 0.5f;                   // 2^-1
  constexpr float m3 = 0.35355339059327376f;   // 2^-1.5
  constexpr float m4 = 0.25f;                  // 2^-2
  constexpr float m5 = 0.17677669529663687f;   // 2^-2.5
  if (r == 0) return m5;                       // cA5 row
  const int p = (r >= 16) ? 4 : (r >= 8) ? 3 : (r >= 4) ? 2 : (r >= 2) ? 1 : 0;
  const int l = 5 - p;                         // wavelet level
  const int j = r - (1 << p);
  const int width = 1 << l;
  const int start = j * width;
  if (c < start || c >= start + width) return 0.0f;
  const float mag = (l == 1) ? m1 : (l == 2) ? m2 : (l == 3) ? m3
                    : (l == 4) ? m4 : m5;
  return (c - start < width / 2) ? mag : -mag;
}

// Per-lane A-fragment table, laid out [t][hi][n][pair] so that for a fixed t
// the 32 lanes of a wave read 8B each, fully contiguous (coalesced b64s).
// top: W rows 0..15 (lane M = n); bot: W rows 16..31.
// 16x4 f32 A layout: lane holds M=lane%16; VGPR0 = K {0|2}, VGPR1 = K {1|3},
// so lane (hi,n) at K-step t needs columns {4t+2hi, 4t+2hi+1}.
struct WTab {
  float top[8][2][16][2];
  float bot[8][2][16][2];
};

constexpr WTab make_wtab() {
  WTab w{};
  for (int t = 0; t < 8; ++t)
    for (int hi = 0; hi < 2; ++hi)
      for (int n = 0; n < 16; ++n) {
        const int c0 = 4 * t + 2 * hi;
        w.top[t][hi][n][0] = haar_w_c(n,      c0);
        w.top[t][hi][n][1] = haar_w_c(n,      c0 + 1);
        w.bot[t][hi][n][0] = haar_w_c(16 + n, c0);
        w.bot[t][hi][n][1] = haar_w_c(16 + n, c0 + 1);
      }
  return w;
}

__constant__ WTab g_wtab = make_wtab();

__global__ __launch_bounds__(256) void haar_wmma_kernel(
    const float* __restrict__ in, float* __restrict__ out) {
  // Double-buffered per-wave-private LDS slices: 8 waves * 2 * 576 floats
  __shared__ __align__(16) float lds[8 * 2 * SLICE];

  const int lane = threadIdx.x & 31;
  const int wave = threadIdx.x >> 5;
  const int hi   = lane >> 4;     // half-wave: selects K pair / M half
  const int n    = lane & 15;     // column (block) index inside the tile
  const int row  = blockIdx.x;    // one (b,c) row of 8192 floats per block

  float* slice0 = &lds[wave * 2 * SLICE];
  float* slice1 = slice0 + SLICE;

  const int task0 = wave * 2;     // 16 tasks of 512 floats per row
  const float* src0 = in + (size_t)row * 8192 + (size_t)task0 * 512;
  const float* src1 = src0 + 512;

  const int nl = lane >> 3;       // block-within-group-of-4 for staging
  const int cc = lane & 7;        // 16B chunk within a 32-float block

  // ---- Stage both tasks (8 async b128 ops), pipelined ----
#if HAVE_ASYNC_LDS
#pragma unroll
  for (int i = 0; i < 4; ++i) {
    const int blkL = i * 4 + nl;
    __builtin_amdgcn_global_load_async_to_lds_b128(
        (g_b128_ptr)(src0 + blkL * 32 + cc * 4),
        (l_b128_ptr)(slice0 + blkL * LDS_STRIDE + cc * 4), 0, 0);
  }
#pragma unroll
  for (int i = 0; i < 4; ++i) {
    const int blkL = i * 4 + nl;
    __builtin_amdgcn_global_load_async_to_lds_b128(
        (g_b128_ptr)(src1 + blkL * 32 + cc * 4),
        (l_b128_ptr)(slice1 + blkL * LDS_STRIDE + cc * 4), 0, 0);
  }
  __builtin_amdgcn_s_wait_asynccnt(4);  // first 4 done (in-order completion)
#else
#pragma unroll
  for (int i = 0; i < 4; ++i) {
    const int blkL = i * 4 + nl;
    const float4 a = *(const float4*)(src0 + blkL * 32 + cc * 4);
    *(float4*)(slice0 + blkL * LDS_STRIDE + cc * 4) = a;
    const float4 b = *(const float4*)(src1 + blkL * 32 + cc * 4);
    *(float4*)(slice1 + blkL * LDS_STRIDE + cc * 4) = b;
  }
  __syncthreads();
#endif

  // ---- Fetch constant A-fragments (coalesced 8B loads, constant-cached) ----
  v2f a_top[8], a_bot[8];
#pragma unroll
  for (int t = 0; t < 8; ++t) {
    a_top[t].x = g_wtab.top[t][hi][n][0];
    a_top[t].y = g_wtab.top[t][hi][n][1];
    a_bot[t].x = g_wtab.bot[t][hi][n][0];
    a_bot[t].y = g_wtab.bot[t][hi][n][1];
  }

  float* orow = out + (size_t)row * 8192;

#pragma unroll
  for (int ti = 0; ti < 2; ++ti) {
    const float* wlds = ti ? slice1 : slice0;
#if HAVE_ASYNC_LDS
    if (ti == 1) __builtin_amdgcn_s_wait_asynccnt(0);  // second buffer ready
#endif

    // ---- Y(32x16) = W(32x32) * X(32x16) via 16 x v_wmma_f32_16x16x4_f32 ----
    // B 4x16 layout: lane holds N=lane%16; VGPR0 = K {0|2}, VGPR1 = K {1|3}.
    v8f accT = {};
    v8f accB = {};
#pragma unroll
    for (int t = 0; t < 8; ++t) {
      const int k0 = 4 * t + 2 * hi;
      v2f b;
      b.x = wlds[n * LDS_STRIDE + k0];
      b.y = wlds[n * LDS_STRIDE + k0 + 1];
      accT = __builtin_amdgcn_wmma_f32_16x16x4_f32(
          false, a_top[t], false, b, (short)0, accT, false, false);
      accB = __builtin_amdgcn_wmma_f32_16x16x4_f32(
          false, a_bot[t], false, b, (short)0, accB, false, false);
    }

    // ---- Scatter, closed-formed per half-wave (D: lane holds rows v+8*hi,
    //      col n). hi=0 -> accT rows 0..7 (cA5,cD5,cD4x2,cD3x4);
    //      hi=1 -> accT rows 8..15 (cD2, consecutive addresses). accB is
    //      always cD1 with consecutive addresses. All wide stores aligned.
    const int blk = (task0 + ti) * 16 + n;   // global 32-block index in row
    if (hi == 0) {
      orow[blk]       = accT[0];                               // cA5
      orow[256 + blk] = accT[1];                               // cD5
      float2 d4 = {accT[2], accT[3]};
      *(float2*)(orow + 512 + 2 * blk) = d4;                   // cD4 (8B align)
      float4 d3 = {accT[4], accT[5], accT[6], accT[7]};
      *(float4*)(orow + 1024 + 4 * blk) = d3;                  // cD3 (16B align)
    } else {
      float4 d2lo = {accT[0], accT[1], accT[2], accT[3]};
      float4 d2hi = {accT[4], accT[5], accT[6], accT[7]};
      *(float4*)(orow + 2048 + 8 * blk)     = d2lo;            // cD2
      *(float4*)(orow + 2048 + 8 * blk + 4) = d2hi;
    }
    float4 d1lo = {accB[0], accB[1], accB[2], accB[3]};
    float4 d1hi = {accB[4], accB[5], accB[6], accB[7]};
    *(float4*)(orow + 4096 + 16 * blk + 8 * hi)     = d1lo;    // cD1
    *(float4*)(orow + 4096 + 16 * blk + 8 * hi + 4) = d1hi;
  }
}

extern "C" void kernel_launch(void* const* d_in, const int* in_sizes, int n_in,
                              void* d_out, int out_size, void* d_ws, size_t ws_size,
                              hipStream_t stream) {
  (void)in_sizes; (void)n_in; (void)d_ws; (void)ws_size;
  const float* x = (const float*)d_in[0];
  float* out = (float*)d_out;
  const int rows = out_size / 8192;  // 64*64 = 4096 rows of length 8192
  if (rows <= 0) return;
  haar_wmma_kernel<<<dim3(rows), dim3(256), 0, stream>>>(x, out);
}